// ObjCondensationLoss_61254823576077
// MI455X (gfx1250) — compile-verified
//
#include <hip/hip_runtime.h>
#include <math.h>

#define MAX_K   1024          // compile-time bound on n_true (actual K read on device)
#define RED_T   256
#define CLS_STRIDE 12         // floats per class row in LDS: 8 xa + xa2 + qak + 2 pad
                              // 48B stride: 16B-aligned for B128, conflict-free (12c mod 64 distinct)

typedef __attribute__((ext_vector_type(2))) float v2f;
typedef __attribute__((ext_vector_type(8))) float v8f;

static __host__ __device__ inline size_t align256(size_t x) { return (x + 255) & ~(size_t)255; }

// LDS byte offset of a generic pointer into __shared__ memory:
// AMDGPU flat-LDS aperture keeps the LDS offset in addr[31:0].
__device__ __forceinline__ unsigned ldsOff(const void* p) {
    return (unsigned)(size_t)p;
}

// ---------------- init per-class state ----------------
__global__ void k_init(unsigned long long* __restrict__ keys,
                       unsigned* __restrict__ fcent) {
    int k = blockIdx.x * blockDim.x + threadIdx.x;
    if (k < MAX_K) { keys[k] = 0ull; fcent[k] = 0u; }
}

// ---------------- per-hit: q, |x|^2, class argmax, background partials ----------------
__global__ void k_hits(const float* __restrict__ x, const float* __restrict__ f,
                       const int* __restrict__ y, int N,
                       float* __restrict__ q, float* __restrict__ x2,
                       unsigned long long* __restrict__ keys,
                       float* __restrict__ pBkg, unsigned* __restrict__ pCnt) {
    __shared__ float    sSum[RED_T];
    __shared__ unsigned sCnt[RED_T];
    int tid = threadIdx.x;
    int i   = blockIdx.x * RED_T + tid;
    float    fb = 0.0f;
    unsigned cb = 0u;
    if (i < N) {
        float fi = f[i];
        float t  = atanhf(fi);
        float qi = t * t + 0.5f;                    // Q_MIN = 0.5
        q[i] = qi;
        float s = 0.0f;
        #pragma unroll
        for (int d = 0; d < 8; ++d) { float v = x[(size_t)i * 8 + d]; s += v * v; }
        x2[i] = s;
        int yi = y[i];
        if (yi >= 0 && yi < MAX_K) {
            // larger q wins; ties -> smaller index (matches jnp.argmax first-occurrence)
            unsigned long long key =
                ((unsigned long long)__float_as_uint(qi) << 32) |
                (unsigned long long)(0xFFFFFFFFu - (unsigned)i);
            atomicMax(&keys[yi], key);
        } else if (yi == -1) {
            fb = fi; cb = 1u;
        }
    }
    sSum[tid] = fb; sCnt[tid] = cb;
    __syncthreads();
    for (int s = RED_T / 2; s > 0; s >>= 1) {
        if (tid < s) { sSum[tid] += sSum[tid + s]; sCnt[tid] += sCnt[tid + s]; }
        __syncthreads();
    }
    if (tid == 0) { pBkg[blockIdx.x] = sSum[0]; pCnt[blockIdx.x] = sCnt[0]; }
}

// ---------------- edges: segment max of f over edge_j ----------------
__global__ void k_edges(const int* __restrict__ ei, const int* __restrict__ ej,
                        const float* __restrict__ f, int E,
                        unsigned* __restrict__ fcent) {
    int e = blockIdx.x * blockDim.x + threadIdx.x;
    if (e < E) {
        int k = ej[e];
        if (k >= 0 && k < MAX_K) {
            float v = f[ei[e]];                 // f > 0 -> float bits monotone as u32
            atomicMax(&fcent[k], __float_as_uint(v));
        }
    }
}

// ---------------- gather condensation points ----------------
__global__ void k_centers(const float* __restrict__ x, const int* __restrict__ nTruePtr,
                          const unsigned long long* __restrict__ keys,
                          float* __restrict__ qak, float* __restrict__ xa2,
                          float* __restrict__ xa) {
    int k = blockIdx.x * blockDim.x + threadIdx.x;
    if (k >= MAX_K) return;
    int K = *nTruePtr;
    if (k < K) {
        unsigned long long key = keys[k];
        float    qv  = __uint_as_float((unsigned)(key >> 32));
        unsigned idx = 0xFFFFFFFFu - (unsigned)(key & 0xFFFFFFFFull);
        if (key == 0ull) { qv = 0.0f; idx = 0u; }   // empty class -> zero weight
        qak[k] = qv;
        float s = 0.0f;
        #pragma unroll
        for (int d = 0; d < 8; ++d) {
            float v = x[(size_t)idx * 8 + d];
            xa[(size_t)k * 8 + d] = v;
            s += v * v;
        }
        xa2[k] = s;
    } else {
        qak[k] = 0.0f; xa2[k] = 0.0f;
        #pragma unroll
        for (int d = 0; d < 8; ++d) xa[(size_t)k * 8 + d] = 0.0f;
    }
}

// ---------------- main: N x K potentials via V_WMMA_F32_16X16X4_F32 ----------------
// One wave per 16-hit tile. Per-class data (x_a rows, |x_a|^2, q_ak) is staged once
// per block into LDS via GLOBAL_LOAD_ASYNC_TO_LDS (ASYNCcnt), then each wave runs
// the k-tile loop out of LDS: A = x rows (16x4 f32, two fragments cover D=8),
// B = x_a^T (4x16), two chained WMMAs -> 16x16 gram tile, hinge epilogue in regs.
#define WPB 8   // waves per block (256 threads)
__global__ void __launch_bounds__(256)
k_main(const float* __restrict__ x, const float* __restrict__ q,
       const float* __restrict__ x2, const int* __restrict__ y,
       const float* __restrict__ xa, const float* __restrict__ xa2,
       const float* __restrict__ qak, const int* __restrict__ nTruePtr,
       int N, int numTiles, float* __restrict__ pLv) {
    __shared__ __align__(16) float sCls[MAX_K * CLS_STRIDE];   // 48 KB

    int tid  = threadIdx.x;
    int lane = tid & 31;
    int w    = blockIdx.x * WPB + (tid >> 5);

    int K      = *nTruePtr;
    int kTiles = (K + 15) >> 4;          // kTiles*16 <= MAX_K (zero-padded by k_centers)
    int Kuse   = kTiles << 4;

    // ---- stage per-class data to LDS with async global->LDS copies ----
    for (int c = tid; c < Kuse; c += 256) {
        unsigned     l = ldsOff(&sCls[c * CLS_STRIDE]);
        const float* g = xa + (size_t)c * 8;
        // INST_OFFSET is added to both the LDS and global addresses (ISA 15.18.3)
        asm volatile("global_load_async_to_lds_b128 %0, %1, off"
                     :: "v"(l), "v"(g) : "memory");
        asm volatile("global_load_async_to_lds_b128 %0, %1, off offset:16"
                     :: "v"(l), "v"(g) : "memory");
        asm volatile("global_load_async_to_lds_b32 %0, %1, off"
                     :: "v"(l + 32u), "v"(xa2 + c) : "memory");
        asm volatile("global_load_async_to_lds_b32 %0, %1, off"
                     :: "v"(l + 36u), "v"(qak + c) : "memory");
    }
    asm volatile("s_wait_asynccnt 0" ::: "memory");
    __syncthreads();

    if (w >= numTiles) return;

    int half = lane >> 4;                // 0: K pair {0,1}/{4,5}; 1: {2,3}/{6,7}
    int mrow = lane & 15;
    int base = w * 16;

    // A fragments: lane holds row M=mrow, K pair (half*2) and (4+half*2)
    int arow = base + mrow; if (arow >= N) arow = N - 1;
    const float* ap = x + (size_t)arow * 8 + (half << 1);
    v2f aLo, aHi;
    aLo.x = ap[0]; aLo.y = ap[1];
    aHi.x = ap[4]; aHi.y = ap[5];

    // Epilogue row data: C layout gives this lane rows M = half*8 + r
    float rx2[8], rq[8]; int ry[8];
    #pragma unroll
    for (int r = 0; r < 8; ++r) {
        int row = base + half * 8 + r;
        if (row < N) { rx2[r] = x2[row]; rq[r] = q[row]; ry[r] = y[row]; }
        else         { rx2[r] = 0.0f;    rq[r] = 0.0f;   ry[r] = -2;    }
    }

    float acc = 0.0f;
    for (int kt = 0; kt < kTiles; ++kt) {
        int col = kt * 16 + mrow;        // this lane's N column (B & C layout)
        const float* bp = &sCls[col * CLS_STRIDE + (half << 1)];
        v2f bLo, bHi;
        bLo.x = bp[0]; bLo.y = bp[1];
        bHi.x = bp[4]; bHi.y = bp[5];

        v8f c = {0.f, 0.f, 0.f, 0.f, 0.f, 0.f, 0.f, 0.f};
        c = __builtin_amdgcn_wmma_f32_16x16x4_f32(false, aLo, false, bLo,
                                                  (short)0, c, false, false);
        c = __builtin_amdgcn_wmma_f32_16x16x4_f32(false, aHi, false, bHi,
                                                  (short)0, c, false, false);

        float x2c = sCls[col * CLS_STRIDE + 8];
        float qc  = sCls[col * CLS_STRIDE + 9];   // 0 for padded columns
        #pragma unroll
        for (int r = 0; r < 8; ++r) {
            float d = rx2[r] + x2c - 2.0f * c[r];
            d = fmaxf(d, 0.0f);
            float term = (ry[r] == col) ? d : fmaxf(1.0f - d, 0.0f);
            acc = fmaf(rq[r] * qc, term, acc);
        }
    }
    // wave32 reduction
    #pragma unroll
    for (int off = 16; off > 0; off >>= 1) acc += __shfl_xor(acc, off, 32);
    if (lane == 0) pLv[w] = acc;
}

// ---------------- final fixed-order reduction & combine ----------------
__global__ void k_final(const unsigned* __restrict__ fcent,
                        const float* __restrict__ pLv, int nLv,
                        const float* __restrict__ pBkg, const unsigned* __restrict__ pCnt, int nB,
                        const int* __restrict__ nTruePtr, int N,
                        float* __restrict__ out) {
    __shared__ float    sA[RED_T], sB[RED_T], sC[RED_T];
    __shared__ unsigned sD[RED_T];
    int tid = threadIdx.x;
    int K   = *nTruePtr;
    float fc = 0.0f, lv = 0.0f, bs = 0.0f; unsigned cnt = 0u;
    for (int k = tid; k < K;   k += RED_T) fc += __uint_as_float(fcent[k]);
    for (int i = tid; i < nLv; i += RED_T) lv += pLv[i];
    for (int i = tid; i < nB;  i += RED_T) { bs += pBkg[i]; cnt += pCnt[i]; }
    sA[tid] = fc; sB[tid] = lv; sC[tid] = bs; sD[tid] = cnt;
    __syncthreads();
    for (int s = RED_T / 2; s > 0; s >>= 1) {
        if (tid < s) { sA[tid] += sA[tid+s]; sB[tid] += sB[tid+s];
                       sC[tid] += sC[tid+s]; sD[tid] += sD[tid+s]; }
        __syncthreads();
    }
    if (tid == 0) {
        float b1 = 1.0f - sA[0] / (float)K;
        float b2 = (sD[0] > 0u) ? (sC[0] / (float)sD[0]) : 0.0f;  // S_B = 1
        float Lv = sB[0] / (float)N;
        out[0] = b1 + b2 + Lv;
    }
}

extern "C" void kernel_launch(void* const* d_in, const int* in_sizes, int n_in,
                              void* d_out, int out_size, void* d_ws, size_t ws_size,
                              hipStream_t stream) {
    const float* x  = (const float*)d_in[0];
    const float* f  = (const float*)d_in[1];
    const int*   y  = (const int*)d_in[2];
    const int*   ei = (const int*)d_in[3];
    const int*   ej = (const int*)d_in[4];
    const int*   nT = (const int*)d_in[5];   // n_true (device scalar)

    const int N = in_sizes[1];
    const int E = in_sizes[3];
    const int numTiles = (N + 15) / 16;
    const int nB       = (N + RED_T - 1) / RED_T;

    // workspace layout (256B-aligned sections)
    char*  ws  = (char*)d_ws;
    size_t off = 0;
    float*              q     = (float*)(ws + off);              off = align256(off + (size_t)N * 4);
    float*              x2    = (float*)(ws + off);              off = align256(off + (size_t)N * 4);
    unsigned long long* keys  = (unsigned long long*)(ws + off); off = align256(off + (size_t)MAX_K * 8);
    unsigned*           fcent = (unsigned*)(ws + off);           off = align256(off + (size_t)MAX_K * 4);
    float*              qak   = (float*)(ws + off);              off = align256(off + (size_t)MAX_K * 4);
    float*              xa2   = (float*)(ws + off);              off = align256(off + (size_t)MAX_K * 4);
    float*              xa    = (float*)(ws + off);              off = align256(off + (size_t)MAX_K * 32);
    float*              pLv   = (float*)(ws + off);              off = align256(off + (size_t)numTiles * 4);
    float*              pBkg  = (float*)(ws + off);              off = align256(off + (size_t)nB * 4);
    unsigned*           pCnt  = (unsigned*)(ws + off);           off = align256(off + (size_t)nB * 4);
    (void)ws_size; (void)n_in; (void)out_size;

    k_init   <<<(MAX_K + 255) / 256, 256, 0, stream>>>(keys, fcent);
    k_hits   <<<nB, RED_T, 0, stream>>>(x, f, y, N, q, x2, keys, pBkg, pCnt);
    k_edges  <<<(E + 255) / 256, 256, 0, stream>>>(ei, ej, f, E, fcent);
    k_centers<<<(MAX_K + 255) / 256, 256, 0, stream>>>(x, nT, keys, qak, xa2, xa);
    k_main   <<<(numTiles + WPB - 1) / WPB, 256, 0, stream>>>(x, q, x2, y, xa, xa2, qak, nT,
                                                              N, numTiles, pLv);
    k_final  <<<1, RED_T, 0, stream>>>(fcent, pLv, numTiles, pBkg, pCnt, nB, nT, N,
                                       (float*)d_out);
}